// GCN_42786464203042
// MI455X (gfx1250) — compile-verified
//
#include <hip/hip_runtime.h>
#include <math.h>

#define B_  32
#define N_  300
#define T_  12
#define D_  64
#define M_  16
#define NP  304   // N padded to 16
#define NTI 19    // 304/16 tiles

// LDS row strides chosen for conflict-free wave32 fragment access
#define TES 67
#define AS  307
#define PS  68

typedef __attribute__((ext_vector_type(2))) float v2f;
typedef __attribute__((ext_vector_type(8))) float v8f;

__device__ __forceinline__ v8f wmma_f32(v2f a, v2f b, v8f c) {
  // D = A(16x4 f32) * B(4x16 f32) + C(16x16 f32)
  return __builtin_amdgcn_wmma_f32_16x16x4_f32(false, a, false, b, (short)0, c, false, false);
}

// out[r,c] = sum_m A[r,m] * B[m,c] (K=16), zero-padded to [rowsP, colsP]
__global__ void k_mm16p(const float* __restrict__ A, const float* __restrict__ Bm,
                        float* __restrict__ out, int rowsV, int colsV,
                        int rowsP, int colsP) {
  int idx = blockIdx.x * blockDim.x + threadIdx.x;
  if (idx >= rowsP * colsP) return;
  int r = idx / colsP, c = idx % colsP;
  float acc = 0.f;
  if (r < rowsV && c < colsV) {
#pragma unroll
    for (int m = 0; m < M_; ++m)
      acc = fmaf(A[r * M_ + m], Bm[m * colsV + c], acc);
  }
  out[idx] = acc;
}

// adjWt[k][n] (D x NP, zero-padded), gcnWt[f][d] (3D x D), adjBp[n] (NP, padded)
__global__ void k_prep(const float* __restrict__ adjW, const float* __restrict__ gcnW,
                       const float* __restrict__ adjB,
                       float* __restrict__ adjWt, float* __restrict__ gcnWt,
                       float* __restrict__ adjBp) {
  int idx = blockIdx.x * blockDim.x + threadIdx.x;
  if (idx < D_ * NP) {
    int k = idx / NP, n = idx % NP;
    adjWt[idx] = (n < N_) ? adjW[n * D_ + k] : 0.f;
  } else if (idx < D_ * NP + 3 * D_ * D_) {
    int j = idx - D_ * NP;
    int f = j / D_, d = j % D_;
    gcnWt[j] = gcnW[d * (3 * D_) + f];
  } else if (idx < D_ * NP + 3 * D_ * D_ + NP) {
    int n = idx - (D_ * NP + 3 * D_ * D_);
    adjBp[n] = (n < N_) ? adjB[n] : 0.f;
  }
}

// H0[bt, n, d] = X[b, d, n, t], rows >= N zeroed
__global__ void k_xp(const float* __restrict__ X, float* __restrict__ H0) {
  size_t idx = (size_t)blockIdx.x * blockDim.x + threadIdx.x;
  if (idx >= (size_t)B_ * T_ * NP * D_) return;
  int d  = (int)(idx & (D_ - 1));
  int n  = (int)((idx >> 6) % NP);
  int bt = (int)(idx / ((size_t)NP * D_));
  int b = bt / T_, t = bt % T_;
  float v = 0.f;
  if (n < N_) v = X[(((size_t)b * D_ + d) * N_ + n) * T_ + t];
  H0[idx] = v;
}

// One workgroup per (b*t, 16-row tile). 128 threads = 4 waves.
__global__ __launch_bounds__(128) void k_hop(
    const float* __restrict__ te,    // [B,N,T,D]
    const float* __restrict__ adjBp, // [NP] padded
    const float* __restrict__ S,     // [NP,NP] padded
    const float* __restrict__ Bmm,   // [NP,NP] padded
    const float* __restrict__ adjWt, // [D,NP]
    const float* __restrict__ Wbuf,  // [NP,D,D] padded rows
    const float* __restrict__ H0,    // [BT,NP,D]
    const float* __restrict__ Hb,    // [BT,NP,D]
    float* __restrict__ Hout)        // [BT,NP,D]
{
  __shared__ float te_s[16 * TES];  // te tile [16][64]
  __shared__ float A_s[16 * AS];    // A row-strip [16][NP]
  __shared__ float P_s[16 * PS];    // (A@Hb) tile [16][64]

  const int bt = blockIdx.x;
  const int b = bt / T_, t = bt % T_;
  const int n0 = blockIdx.y * 16;
  const int tid = threadIdx.x;
  const int wave = tid >> 5;
  const int lane = tid & 31;
  const int half = lane >> 4;       // K-pair / M+8 selector
  const int nl = lane & 15;

  // ---- Stage te tile to LDS (branchless clamp + select, zero pad rows)
  for (int i = tid; i < 16 * 16; i += 128) {
    const int m = i >> 4;
    const int c4 = (i & 15) * 4;
    const int n = n0 + m;
    const int nc = (n < N_) ? n : (N_ - 1);
    const float msk = (n < N_) ? 1.f : 0.f;
    float4 v = *(const float4*)(te + (((size_t)b * N_ + nc) * T_ + t) * D_ + c4);
    te_s[m * TES + c4 + 0] = v.x * msk;
    te_s[m * TES + c4 + 1] = v.y * msk;
    te_s[m * TES + c4 + 2] = v.z * msk;
    te_s[m * TES + c4 + 3] = v.w * msk;
  }
  __syncthreads();

  // ---- Step 1: A[n0:n0+16, :] = relu(S * sin(te . adjW^T + adjB) + Bm)
  for (int cb = wave; cb < NTI; cb += 4) {
    const int c0 = cb * 16;
    const float* bp = adjWt + c0 + nl;
    v8f acc0 = {}, acc1 = {};
#pragma unroll
    for (int kk = 0; kk < D_; kk += 8) {
      const int ka = kk + 2 * half;
      v2f a, bb;
      a.x = te_s[nl * TES + ka + 0];
      a.y = te_s[nl * TES + ka + 1];
      bb.x = bp[(ka + 0) * NP];
      bb.y = bp[(ka + 1) * NP];
      acc0 = wmma_f32(a, bb, acc0);
      a.x = te_s[nl * TES + ka + 4];
      a.y = te_s[nl * TES + ka + 5];
      bb.x = bp[(ka + 4) * NP];
      bb.y = bp[(ka + 5) * NP];
      acc1 = wmma_f32(a, bb, acc1);
    }
    const int col = c0 + nl;
    const float ab = adjBp[col];
#pragma unroll
    for (int v = 0; v < 8; ++v) {
      const int m = v + 8 * half;
      const int n = n0 + m;
      float x = __sinf(acc0[v] + acc1[v] + ab);   // hardware v_sin, branch-free
      x = fmaf(x, S[n * NP + col], Bmm[n * NP + col]);
      A_s[m * AS + col] = fmaxf(x, 0.f);
    }
  }
  __syncthreads();

  // ---- Step 2: P[16,64] = A_tile[16,NP] @ Hb[NP,64] (pad rows of Hb are zero)
  {
    const int d0 = wave * 16;
    const float* hb = Hb + (size_t)bt * NP * D_ + d0 + nl;
    v8f acc0 = {}, acc1 = {};
    for (int k0 = 0; k0 < NP; k0 += 16) {
      if (k0 + 16 < NP) __builtin_prefetch(hb + (k0 + 16) * D_, 0, 3);
#pragma unroll
      for (int kq = 0; kq < 16; kq += 8) {
        const int ka = k0 + kq + 2 * half;
        v2f a, bb;
        a.x = A_s[nl * AS + ka + 0];
        a.y = A_s[nl * AS + ka + 1];
        bb.x = hb[(ka + 0) * D_];
        bb.y = hb[(ka + 1) * D_];
        acc0 = wmma_f32(a, bb, acc0);
        a.x = A_s[nl * AS + ka + 4];
        a.y = A_s[nl * AS + ka + 5];
        bb.x = hb[(ka + 4) * D_];
        bb.y = hb[(ka + 5) * D_];
        acc1 = wmma_f32(a, bb, acc1);
      }
    }
#pragma unroll
    for (int v = 0; v < 8; ++v)
      P_s[(v + 8 * half) * PS + d0 + nl] = acc0[v] + acc1[v];
  }
  __syncthreads();

  // ---- Step 3+4: per-node matmul + gated residual (branch-free; pad W rows = 0)
  for (int i = tid; i < 16 * D_; i += 128) {
    const int m = i >> 6;
    const int d = i & (D_ - 1);
    const int n = n0 + m;
    const float* wrow = Wbuf + (size_t)n * (D_ * D_) + d;
    float acc = 0.f;
#pragma unroll 8
    for (int k = 0; k < D_; ++k)
      acc = fmaf(P_s[m * PS + k], wrow[k * D_], acc);
    const float pre = H0[((size_t)bt * NP + n) * D_ + d] + acc;
    // sigmoid(pre) * tanh(pre), branch-free via fast exp (saturates via IEEE inf)
    const float sg = 1.f / (1.f + __expf(-pre));
    const float th = 1.f - 2.f / (__expf(2.f * pre) + 1.f);
    Hout[((size_t)bt * NP + n) * D_ + d] = sg * th;
  }
}

// Output GEMM: [16,192] concat(H0,H1,H2) tile @ gcnW^T[192,64] + bias, transposed store
__global__ __launch_bounds__(128) void k_gcn(
    const float* __restrict__ H0, const float* __restrict__ H1,
    const float* __restrict__ H2, const float* __restrict__ gcnWt,
    const float* __restrict__ gcnB, float* __restrict__ out)
{
  const int bt = blockIdx.x;
  const int b = bt / T_, t = bt % T_;
  const int n0 = blockIdx.y * 16;
  const int lane = threadIdx.x & 31;
  const int wave = threadIdx.x >> 5;
  const int half = lane >> 4;
  const int nl = lane & 15;
  const int d0 = wave * 16;

  const float* Hs[3] = { H0, H1, H2 };
  v8f acc0 = {}, acc1 = {};
  for (int h = 0; h < 3; ++h) {
    const float* hp = Hs[h] + ((size_t)bt * NP + n0) * D_ + nl * D_;
    const float* wp = gcnWt + (size_t)h * D_ * D_ + d0 + nl;
#pragma unroll
    for (int kk = 0; kk < D_; kk += 8) {
      const int ka = kk + 2 * half;
      v2f a, bb;
      a.x = hp[ka + 0];
      a.y = hp[ka + 1];
      bb.x = wp[(ka + 0) * D_];
      bb.y = wp[(ka + 1) * D_];
      acc0 = wmma_f32(a, bb, acc0);
      a.x = hp[ka + 4];
      a.y = hp[ka + 5];
      bb.x = wp[(ka + 4) * D_];
      bb.y = wp[(ka + 5) * D_];
      acc1 = wmma_f32(a, bb, acc1);
    }
  }
#pragma unroll
  for (int v = 0; v < 8; ++v) {
    const int m = v + 8 * half;
    const int n = n0 + m;
    const int d = d0 + nl;
    if (n < N_)
      out[(((size_t)b * D_ + d) * N_ + n) * T_ + t] = acc0[v] + acc1[v] + gcnB[d];
  }
}

extern "C" void kernel_launch(void* const* d_in, const int* in_sizes, int n_in,
                              void* d_out, int out_size, void* d_ws, size_t ws_size,
                              hipStream_t stream) {
  const float* X      = (const float*)d_in[0];
  const float* te     = (const float*)d_in[1];
  const float* adjW   = (const float*)d_in[2];
  const float* adjB   = (const float*)d_in[3];
  const float* tW1    = (const float*)d_in[4];
  const float* tW2    = (const float*)d_in[5];
  const float* a1     = (const float*)d_in[6];
  const float* a2     = (const float*)d_in[7];
  const float* b1     = (const float*)d_in[8];
  const float* b2     = (const float*)d_in[9];
  const float* gcnW   = (const float*)d_in[10];
  const float* gcnB   = (const float*)d_in[11];
  float* out = (float*)d_out;

  float* ws    = (float*)d_ws;
  float* Wbuf  = ws;                  // NP*D*D     = 1245184
  float* S     = Wbuf + 1245184;      // NP*NP      = 92416
  float* Bmm   = S + 92416;           // NP*NP      = 92416
  float* adjWt = Bmm + 92416;         // D*NP       = 19456
  float* gcnWt = adjWt + 19456;       // 3D*D       = 12288
  float* adjBp = gcnWt + 12288;       // NP (pad to 320)
  float* H0    = adjBp + 320;         // BT*NP*D    = 7471104
  float* H1    = H0 + 7471104;
  float* H2    = H1 + 7471104;

  // Precompute small weight products (zero-padded) + transposed weights
  k_mm16p<<<(NP * NP + 255) / 256, 256, 0, stream>>>(a1, a2, S, N_, N_, NP, NP);
  k_mm16p<<<(NP * NP + 255) / 256, 256, 0, stream>>>(b1, b2, Bmm, N_, N_, NP, NP);
  k_mm16p<<<(NP * D_ * D_ + 255) / 256, 256, 0, stream>>>(tW1, tW2, Wbuf, N_, D_ * D_, NP, D_ * D_);
  k_prep<<<(D_ * NP + 3 * D_ * D_ + NP + 255) / 256, 256, 0, stream>>>(adjW, gcnW, adjB, adjWt, gcnWt, adjBp);
  k_xp<<<(B_ * T_ * NP * D_ + 255) / 256, 256, 0, stream>>>(X, H0);

  dim3 grid(B_ * T_, NTI);
  // hop 1: H0 -> H1 ; hop 2: H1 -> H2
  k_hop<<<grid, 128, 0, stream>>>(te, adjBp, S, Bmm, adjWt, Wbuf, H0, H0, H1);
  k_hop<<<grid, 128, 0, stream>>>(te, adjBp, S, Bmm, adjWt, Wbuf, H0, H1, H2);
  // output linear + transpose
  k_gcn<<<grid, 128, 0, stream>>>(H0, H1, H2, gcnWt, gcnB, out);
}